// ProposalLayer_40175124087347
// MI455X (gfx1250) — compile-verified
//
#include <hip/hip_runtime.h>
#include <hip/hip_bf16.h>
#include <cstdint>
#include <cstddef>

#define B_      4
#define N_      261888
#define P_      6000
#define NPROP_  2000
#define W_      188            /* ceil(P/32) mask words per row   */
#define PPAD_   (W_ * 32)      /* 6016                             */
#define CAP_    8192           /* candidate capacity (pow2, >= P)  */
#define HBINS_  8192           /* histogram bins (13 key bits)     */

// ---------------------------------------------------------------------------
// Kernel 1: per-batch top-P selection, 3 passes over the data:
//   pass 1: 8192-bin LDS histogram of key>>19  (key = float bits; scores are
//           uniform [0,1) => positive => uint compare order-preserving)
//   pass 2: 8192-bin LDS histogram of bits [18:6] within the threshold bin
//   pass 3: compact all keys >= 26-bit-resolved threshold T into LDS
// then an in-LDS 8192-wide bitonic sort of (key<<32 | ~index) => score desc,
// index-asc tie-break (matches jax.lax.top_k stability). 64KB static LDS;
// histogram (32KB) + reduction scratch (4KB) alias the sort buffer.
// ---------------------------------------------------------------------------
__global__ __launch_bounds__(1024) void topk_kernel(const float* __restrict__ rpn_class,
                                                    unsigned* __restrict__ topk_ids)
{
    __shared__ unsigned long long s_sort[CAP_];   // 64 KB
    __shared__ unsigned s_c;
    unsigned* s_hist = (unsigned*)s_sort;                 // bins 0..8191 (32 KB)
    unsigned* s_red  = (unsigned*)(s_sort + HBINS_ / 2);  // next 4 KB (no overlap)

    const int b = blockIdx.x;
    const int t = threadIdx.x;
    const float* sc = rpn_class + (size_t)b * N_ * 2;     // score = [n*2 + 1]

    // ---- pass 1: coarse histogram (top 13 bits) ----
    for (int i = t; i < HBINS_; i += 1024) s_hist[i] = 0u;
    __syncthreads();
    for (int n = t; n < N_; n += 1024) {
        __builtin_prefetch(sc + 2 * (n + 8192) + 1, 0, 0);     // global_prefetch_b8
        const unsigned k = __float_as_uint(sc[2 * n + 1]);
        atomicAdd(&s_hist[k >> 19], 1u);
    }
    __syncthreads();

    // binary search on bins: largest b1 with suffix-count(bin >= b1) >= P
    unsigned lo = 0, hi = HBINS_;                 // g(0)=N>=P, g(HBINS)=0<P
    while (lo + 1 < hi) {
        const unsigned mid = (lo + hi) >> 1;
        unsigned cnt = 0;
        for (int i = t; i < HBINS_; i += 1024)
            if ((unsigned)i >= mid) cnt += s_hist[i];
        s_red[t] = cnt; __syncthreads();
        for (int s = 512; s > 0; s >>= 1) { if (t < s) s_red[t] += s_red[t + s]; __syncthreads(); }
        const unsigned total = s_red[0]; __syncthreads();
        if (total >= (unsigned)P_) lo = mid; else hi = mid;
    }
    const unsigned b1 = lo;
    {   // above1 = count of keys in bins strictly above b1  (< P)
        unsigned cnt = 0;
        for (int i = t; i < HBINS_; i += 1024)
            if ((unsigned)i > b1) cnt += s_hist[i];
        s_red[t] = cnt; __syncthreads();
        for (int s = 512; s > 0; s >>= 1) { if (t < s) s_red[t] += s_red[t + s]; __syncthreads(); }
    }
    const unsigned above1 = s_red[0];
    __syncthreads();

    // ---- pass 2: refine within bin b1 (bits 18..6) ----
    for (int i = t; i < HBINS_; i += 1024) s_hist[i] = 0u;
    __syncthreads();
    for (int n = t; n < N_; n += 1024) {
        const unsigned k = __float_as_uint(sc[2 * n + 1]);
        if ((k >> 19) == b1) atomicAdd(&s_hist[(k >> 6) & (HBINS_ - 1u)], 1u);
    }
    __syncthreads();
    lo = 0; hi = HBINS_;              // above1+g2(0)>=P, above1+g2(HBINS)=above1<P
    while (lo + 1 < hi) {
        const unsigned mid = (lo + hi) >> 1;
        unsigned cnt = 0;
        for (int i = t; i < HBINS_; i += 1024)
            if ((unsigned)i >= mid) cnt += s_hist[i];
        s_red[t] = cnt; __syncthreads();
        for (int s = 512; s > 0; s >>= 1) { if (t < s) s_red[t] += s_red[t + s]; __syncthreads(); }
        const unsigned total = above1 + s_red[0]; __syncthreads();
        if (total >= (unsigned)P_) lo = mid; else hi = mid;
    }
    const unsigned T = (b1 << 19) | (lo << 6);   // count(>=T) >= P, count(>=T+64) < P

    // ---- pass 3: compact candidates (key >= T) into LDS ----
    if (t == 0) s_c = 0u;
    __syncthreads();
    for (int n = t; n < N_; n += 1024) {
        const unsigned k = __float_as_uint(sc[2 * n + 1]);
        if (k >= T) {
            const unsigned slot = atomicAdd(&s_c, 1u);
            if (slot < CAP_)
                s_sort[slot] = ((unsigned long long)k << 32) |
                               (unsigned long long)(0xFFFFFFFFu - (unsigned)n);
        }
    }
    __syncthreads();
    unsigned cnt = s_c; if (cnt > CAP_) cnt = CAP_;
    for (unsigned i = cnt + t; i < CAP_; i += 1024) s_sort[i] = 0ull;  // pad sorts last
    __syncthreads();

    // ---- bitonic sort, descending ----
    for (unsigned ksz = 2; ksz <= CAP_; ksz <<= 1) {
        for (unsigned j = ksz >> 1; j > 0; j >>= 1) {
            for (unsigned idx = t; idx < CAP_; idx += 1024) {
                const unsigned pr = idx ^ j;
                if (pr > idx) {
                    const bool desc = ((idx & ksz) == 0);
                    const unsigned long long a = s_sort[idx];
                    const unsigned long long c = s_sort[pr];
                    const bool sw = desc ? (a < c) : (a > c);
                    if (sw) { s_sort[idx] = c; s_sort[pr] = a; }
                }
            }
            __syncthreads();
        }
    }

    for (int i = t; i < P_; i += 1024) {
        const unsigned long long v = s_sort[i];
        const unsigned id = (v == 0ull) ? 0u : (0xFFFFFFFFu - (unsigned)(v & 0xFFFFFFFFu));
        topk_ids[(size_t)b * P_ + i] = id;
    }
}

// ---------------------------------------------------------------------------
// Kernel 2: gather + delta decode + clip + normalize -> boxes[b][p] (float4)
// ---------------------------------------------------------------------------
__global__ void decode_kernel(const float* __restrict__ rpn_bbox,
                              const float* __restrict__ anchors,
                              const unsigned* __restrict__ topk_ids,
                              float4* __restrict__ boxes)
{
    const int i = blockIdx.x * blockDim.x + threadIdx.x;
    if (i >= B_ * P_) return;
    const int b = i / P_;
    const unsigned n = topk_ids[i];
    const float* d = rpn_bbox + (((size_t)b * N_ + n) << 2);
    const float* a = anchors + ((size_t)n << 2);

    const float dy = d[0] * 0.1f, dx = d[1] * 0.1f;
    const float dh = d[2] * 0.2f, dw = d[3] * 0.2f;
    const float a0 = a[0], a1 = a[1], a2 = a[2], a3 = a[3];
    float h = a2 - a0, w = a3 - a1;
    const float cy = a0 + 0.5f * h + dy * h;
    const float cx = a1 + 0.5f * w + dx * w;
    h *= __expf(dh);
    w *= __expf(dw);

    const float inv = 1.0f / 1024.0f;
    const float y1 = fminf(fmaxf(cy - 0.5f * h, 0.0f), 1024.0f) * inv;
    const float x1 = fminf(fmaxf(cx - 0.5f * w, 0.0f), 1024.0f) * inv;
    const float y2 = fminf(fmaxf(cy + 0.5f * h, 0.0f), 1024.0f) * inv;
    const float x2 = fminf(fmaxf(cx + 0.5f * w, 0.0f), 1024.0f) * inv;
    boxes[i] = make_float4(y1, x1, y2, x2);
}

// ---------------------------------------------------------------------------
// Kernel 3: NMS suppression bitmask. 256-thread (8-wave) blocks: 256 rows x
// 32 cols per block. Wave 0 stages the 32 column boxes into LDS with the
// gfx1250 async global->LDS copy; all 8 waves reuse the tile.
// mask[b][i][colBlk] bit jj = (iou(i, colBlk*32+jj) > 0.7) && (j > i)
// ---------------------------------------------------------------------------
__global__ __launch_bounds__(256) void nms_mask_kernel(const float4* __restrict__ boxes,
                                                       unsigned* __restrict__ masks)
{
    const int colBlk  = blockIdx.x;           // 0..W_-1
    const int rowBase = blockIdx.y * 256;
    const int b       = blockIdx.z;
    const int t       = threadIdx.x;
    const int i       = rowBase + t;
    const int c0      = colBlk * 32;

    if (c0 + 31 < rowBase) {                  // entire tile strictly below diagonal
        if (i < P_) masks[((size_t)b * PPAD_ + i) * W_ + colBlk] = 0u;
        return;
    }

    __shared__ float4 s_col[32];
    const float4* bx = boxes + (size_t)b * P_;

    if (t < 32) {   // wave 0: async stage 32 column boxes (16B each) into LDS
        const int j  = c0 + t;
        const int jc = (j < P_) ? j : (P_ - 1);
        const unsigned lds_off = (unsigned)(size_t)(&s_col[t]);   // low 32b = LDS offset
        const unsigned voff    = (unsigned)jc * 16u;
        const unsigned long long sbase = (unsigned long long)(const void*)bx;
        asm volatile("global_load_async_to_lds_b128 %0, %1, %2"
                     :: "v"(lds_off), "v"(voff), "s"(sbase)
                     : "memory");
        asm volatile("s_wait_asynccnt 0x0" ::: "memory");
    }
    __syncthreads();

    if (i >= P_) return;
    const float4 rb = bx[i];
    const float areaI = (rb.z - rb.x) * (rb.w - rb.y);

    unsigned m = 0u;
    #pragma unroll 8
    for (int jj = 0; jj < 32; ++jj) {
        const int jg = c0 + jj;
        if (jg >= P_ || jg <= i) continue;
        const float4 cb = s_col[jj];
        const float iy1 = fmaxf(rb.x, cb.x);
        const float ix1 = fmaxf(rb.y, cb.y);
        const float iy2 = fminf(rb.z, cb.z);
        const float ix2 = fminf(rb.w, cb.w);
        const float inter = fmaxf(iy2 - iy1, 0.0f) * fmaxf(ix2 - ix1, 0.0f);
        const float areaJ = (cb.z - cb.x) * (cb.w - cb.y);
        const float uni   = fmaxf(areaI + areaJ - inter, 1e-8f);
        if (inter > 0.7f * uni) m |= (1u << jj);
    }
    masks[((size_t)b * PPAD_ + i) * W_ + colBlk] = m;
}

// ---------------------------------------------------------------------------
// Kernel 4: serial greedy NMS scan + emit first NPROP survivors.
// One block per batch; 188-word removal bitmap in LDS; next mask row is
// prefetched each iteration to hide L2 latency behind the barriers.
// ---------------------------------------------------------------------------
__global__ __launch_bounds__(256) void nms_out_kernel(const float4* __restrict__ boxes,
                                                      const unsigned* __restrict__ masks,
                                                      float* __restrict__ out)
{
    __shared__ unsigned s_remv[W_];
    __shared__ int s_keep;
    __shared__ int s_cnt;

    const int b = blockIdx.x;
    const int t = threadIdx.x;
    const unsigned* mb = masks + (size_t)b * PPAD_ * W_;
    const float4*   bx = boxes + (size_t)b * P_;
    float*          ob = out + (size_t)b * NPROP_ * 4;

    if (t < W_) s_remv[t] = 0u;
    if (t == 0) s_cnt = 0;
    for (int i = t; i < NPROP_ * 4; i += 256) ob[i] = 0.0f;   // d_out is poisoned
    __syncthreads();

    for (int i = 0; i < P_; ++i) {
        if (t < W_ && i + 1 < P_)
            __builtin_prefetch(&mb[(size_t)(i + 1) * W_ + t], 0, 0);  // speculative row i+1
        if (t == 0)
            s_keep = (((s_remv[i >> 5] >> (i & 31)) & 1u) == 0u) ? 1 : 0;
        __syncthreads();
        if (s_keep) {
            if (t < W_) s_remv[t] |= mb[(size_t)i * W_ + t];
            if (t == 0) {
                const int c = s_cnt;
                if (c < NPROP_) {
                    const float4 v = bx[i];
                    ob[c * 4 + 0] = v.x; ob[c * 4 + 1] = v.y;
                    ob[c * 4 + 2] = v.z; ob[c * 4 + 3] = v.w;
                }
                s_cnt = c + 1;
            }
        }
        __syncthreads();
        if (s_cnt >= NPROP_) break;    // uniform read after barrier
    }
}

// ---------------------------------------------------------------------------
extern "C" void kernel_launch(void* const* d_in, const int* in_sizes, int n_in,
                              void* d_out, int out_size, void* d_ws, size_t ws_size,
                              hipStream_t stream)
{
    const float* rpn_class = (const float*)d_in[0];   // (B, N, 2)
    const float* rpn_bbox  = (const float*)d_in[1];   // (B, N, 4)
    const float* anchors   = (const float*)d_in[2];   // (N, 4)
    float* out = (float*)d_out;                       // (B, 2000, 4)

    char* ws = (char*)d_ws;
    unsigned* topk_ids = (unsigned*)ws;                                  // B*P*4   = 96000 B
    float4*   boxes    = (float4*)(ws + 96000);                          // B*P*16  = 384000 B
    unsigned* masks    = (unsigned*)(ws + 96000 + (size_t)B_ * P_ * 16); // B*PPAD*W*4 ~ 18.1 MB

    topk_kernel<<<B_, 1024, 0, stream>>>(rpn_class, topk_ids);
    decode_kernel<<<(B_ * P_ + 255) / 256, 256, 0, stream>>>(rpn_bbox, anchors, topk_ids, boxes);
    nms_mask_kernel<<<dim3(W_, (PPAD_ + 255) / 256, B_), 256, 0, stream>>>(boxes, masks);
    nms_out_kernel<<<B_, 256, 0, stream>>>(boxes, masks, out);
}